// MultivariateDiagonalGaussian_5720896438509
// MI455X (gfx1250) — compile-verified
//
#include <hip/hip_runtime.h>
#include <hip/hip_bf16.h>
#include <stdint.h>

// Problem size (fixed by the reference).
#define N 8192
#define ROWS_PER_BLOCK 16                 // rows streamed per TDM descriptor (iterate_count = 15)
#define FILL_BLOCKS (N / ROWS_PER_BLOCK)  // 512 workgroups
#define NTHREADS 256

typedef uint32_t u32x4 __attribute__((ext_vector_type(4)));
typedef uint32_t u32x8 __attribute__((ext_vector_type(8)));

// ---------------------------------------------------------------------------
// Single fused kernel.
// Per block (512 blocks, each owning 16 output rows):
//   1) Redundantly compute scalar = -N^2/2 - N/2 + sum(log(sigma)).
//      Fixed-order per-thread accumulation + LDS tree => every block (and
//      every call) produces the bit-identical value. ~32 KiB read, L2-hot.
//   2) Fill a 32 KiB LDS row image with `scalar`.
//   3) Wave 0 issues ONE tensor_store_from_lds whose descriptor iterates 16x
//      with lds_addr_increment = 0 (replay the same LDS row) and
//      global_addr_increment = N elements (advance one output row per
//      iteration): 512 KiB of HBM stores per DMA descriptor, NT so the
//      256 MiB stream doesn't thrash the 192 MiB L2, DEV scope so the
//      tensorcnt wait implies device-scope completion.
//   4) After s_wait_tensorcnt 0, lanes 0..15 overwrite their 16 diagonal
//      elements: out[r*N+r] = scalar - 0.5*d*d/sigma.
// ---------------------------------------------------------------------------
__global__ void mdg_fused(const float* __restrict__ values,
                          const float* __restrict__ mu,
                          const float* __restrict__ sigma,
                          float* __restrict__ out) {
    __shared__ float row[N];        // 32 KiB row image
    __shared__ float red[NTHREADS]; // reduction scratch

    // ---- 1) deterministic block-local reduction: sum(log(sigma)) ----------
    float acc = 0.0f;
    for (int i = threadIdx.x; i < N; i += NTHREADS) {
        acc += logf(sigma[i]);
    }
    red[threadIdx.x] = acc;
    __syncthreads();
    for (int s = NTHREADS / 2; s > 0; s >>= 1) {
        if ((int)threadIdx.x < s) red[threadIdx.x] += red[threadIdx.x + s];
        __syncthreads();
    }
    const float n = (float)N;
    const float scalar = -n * n * 0.5f - n * 0.5f + red[0];
    __syncthreads();  // red[0] consumed by all before any reuse of LDS timing

    // ---- 2) stage one scalar-filled output row in LDS ---------------------
    for (int i = threadIdx.x; i < N; i += NTHREADS) {
        row[i] = scalar;
    }
    __syncthreads();  // LDS writes complete & visible before the TDM reads them

    // ---- 3) one TDM descriptor streams 16 rows (512 KiB) to HBM -----------
    if (threadIdx.x < 32) {  // wave 0 only (wave32): exactly one DMA per block
        const uint64_t gaddr =
            (uint64_t)(uintptr_t)(out + (uint64_t)blockIdx.x * (uint64_t)ROWS_PER_BLOCK * (uint64_t)N);
        // Flat shared address: aperture in [63:32], LDS byte offset in [31:0].
        const uint32_t lds_addr = (uint32_t)(uintptr_t)&row[0];

        // D# group 0 (128b): count=1 | lds_addr | global_addr[56:0] | type=2
        u32x4 g0;
        g0[0] = 0x1u;                                   // count=1 (valid user descriptor)
        g0[1] = lds_addr;                               // [63:32] lds_addr (bytes)
        g0[2] = (uint32_t)gaddr;                        // [95:64] global_addr[31:0]
        g0[3] = ((uint32_t)(gaddr >> 32) & 0x01FFFFFFu) // [120:96] global_addr[56:32]
                | 0x80000000u;                          // [127:126] type=2 ("image")

        // D# group 1 (256b)
        u32x8 g1;
        g1[0] = (2u << 16)      // data_size = 2 -> 4 bytes
              | (1u << 19);     // iterate_enable = 1 (workgroup_mask=0: not in cluster)
        g1[1] = ((uint32_t)N & 0xFFFFu) << 16;          // tensor_dim0[15:0]  @ bits [63:48]
        g1[2] = (((uint32_t)N >> 16) & 0xFFFFu)         // tensor_dim0[31:16] @ bits [79:64]
              | (((uint32_t)N & 0xFFFFu) << 16);        // tensor_dim1[15:0]  @ bits [95:80]
        g1[3] = (((uint32_t)N >> 16) & 0xFFFFu)         // tensor_dim1[31:16]
              | (((uint32_t)N & 0xFFFFu) << 16);        // tile_dim0 = N @ bits [127:112]
        g1[4] = 1u;                                     // tile_dim1 = 1, tile_dim2 = 0
        g1[5] = (uint32_t)N;                            // tensor_dim0_stride[31:0] = N
        g1[6] = 0u;                                     // stride0[47:32]=0, dim1_stride[15:0]=0
        g1[7] = 0u;                                     // tensor_dim1_stride hi (unused, 2D tile)

        // D# group 2 (128b), iterate mode
        u32x4 g2;
        g2[0] = 0u;                                     // tensor_dim2 (unused: tile_dim2=0)
        g2[1] = 0u;                                     // lds_addr_increment = 0 (replay same row)
        g2[2] = (uint32_t)N;                            // global_addr_increment[31:0] = N elements
        g2[3] = ((uint32_t)(ROWS_PER_BLOCK - 1)) << 16; // inc[47:32]=0 | iterate_count @ [127:112]

        // D# group 3 (128b): dims 3/4 unused
        u32x4 g3;
        g3[0] = 0u; g3[1] = 0u; g3[2] = 0u; g3[3] = 0u;

        asm volatile("tensor_store_from_lds %0, %1, %2, %3 th:TH_STORE_NT scope:SCOPE_DEV"
                     :
                     : "s"(g0), "s"(g1), "s"(g2), "s"(g3)
                     : "memory");
        __builtin_amdgcn_s_wait_tensorcnt(0);  // DMA complete @ DEV scope; LDS reusable

        // ---- 4) patch this block's 16 diagonal elements -------------------
        if (threadIdx.x < ROWS_PER_BLOCK) {
            const int r = blockIdx.x * ROWS_PER_BLOCK + (int)threadIdx.x;
            const float d = values[r] - mu[r];
            out[(size_t)r * (size_t)N + (size_t)r] = scalar - 0.5f * d * d / sigma[r];
        }
    }
    __syncthreads();  // keep LDS allocated (all waves resident) until DMA completed
}

extern "C" void kernel_launch(void* const* d_in, const int* in_sizes, int n_in,
                              void* d_out, int out_size, void* d_ws, size_t ws_size,
                              hipStream_t stream) {
    (void)in_sizes; (void)n_in; (void)out_size; (void)d_ws; (void)ws_size;

    const float* values = (const float*)d_in[0];
    const float* mu     = (const float*)d_in[1];
    const float* sigma  = (const float*)d_in[2];
    float*       out    = (float*)d_out;

    mdg_fused<<<FILL_BLOCKS, NTHREADS, 0, stream>>>(values, mu, sigma, out);
}